// OrientedSetCriterion_4501125726743
// MI455X (gfx1250) — compile-verified
//
#include <hip/hip_runtime.h>
#include <math.h>

#define QN 900
#define QPAD 912          // 57 full 16-row tiles; rows 900..911 are write-only padding
#define MN 128
#define KCLS 80
#define KP1 81
#define CLS_W 2.0f
#define BBOX_W 5.0f
#define ANG_W 2.0f
#define NOOBJ_W 0.1f
#define NTHREADS 256
#define BIGF 3.0e38f

typedef float v2f __attribute__((ext_vector_type(2)));
typedef float v8f __attribute__((ext_vector_type(8)));

__device__ __forceinline__ float block_sum(float v, float* s, int tid) {
  s[tid] = v;
  __syncthreads();
  for (int off = NTHREADS / 2; off > 0; off >>= 1) {
    if (tid < off) s[tid] += s[tid + off];
    __syncthreads();
  }
  float r = s[0];
  __syncthreads();
  return r;
}

__global__ __launch_bounds__(NTHREADS)
void oriented_criterion_kernel(const float* __restrict__ logits,  // [B,Q,81]
                               const float* __restrict__ pboxes,  // [B,Q,5]
                               const float* __restrict__ tboxes,  // [B,M,5]
                               const float* __restrict__ sizes,   // [B,2]
                               const int*   __restrict__ labels,  // [B,M]
                               float* __restrict__ cost_ws,       // [B,QPAD,M]
                               float* __restrict__ loss_ws)       // [B,3]
{
  const int img  = blockIdx.x;
  const int tid  = threadIdx.x;
  const int lane = tid & 31;
  const int wave = tid >> 5;

  const float* lg = logits + (size_t)img * QN * KP1;
  const float* pb = pboxes + (size_t)img * QN * 5;
  const float* tb = tboxes + (size_t)img * MN * 5;
  const int*   lb = labels + (size_t)img * MN;
  float* cost = cost_ws + (size_t)img * QPAD * MN;

  __shared__ float    s_lse[QN];
  __shared__ float    s_pc[QN], s_ps[QN];     // cos/sin of pred theta
  __shared__ float    s_nb[MN * 5];           // normalized target boxes
  __shared__ float    s_tc[MN], s_ts[MN];     // cos/sin of target theta
  __shared__ int      s_lab[MN];
  __shared__ float    s_rowmin[QN];
  __shared__ int      s_rowarg[QN];
  __shared__ int      s_tgt[QN];
  __shared__ unsigned s_used[4];
  __shared__ int      s_pq[MN], s_pm[MN];
  __shared__ float    s_rv[NTHREADS];
  __shared__ int      s_ri[NTHREADS];

  const float Hh = sizes[img * 2 + 0];
  const float Ww = sizes[img * 2 + 1];

  // ---- stage normalized target boxes, labels, target cos/sin ----
  if (tid < MN) {
    s_nb[tid * 5 + 0] = tb[tid * 5 + 0] / Ww;
    s_nb[tid * 5 + 1] = tb[tid * 5 + 1] / Hh;
    s_nb[tid * 5 + 2] = tb[tid * 5 + 2] / Ww;
    s_nb[tid * 5 + 3] = tb[tid * 5 + 3] / Hh;
    const float th = tb[tid * 5 + 4];
    s_nb[tid * 5 + 4] = th;
    s_tc[tid] = __cosf(th);
    s_ts[tid] = __sinf(th);
    s_lab[tid] = lb[tid];
  }
  if (tid < 4) s_used[tid] = 0u;

  // ---- Phase A: per-query logsumexp + pred theta cos/sin ----
  for (int q = tid; q < QN; q += NTHREADS) {
    const float* row = lg + q * KP1;
    float mx = -BIGF;
    for (int k = 0; k < KP1; ++k) mx = fmaxf(mx, row[k]);
    float s = 0.f;
    for (int k = 0; k < KP1; ++k) s += __expf(row[k] - mx);
    s_lse[q] = mx + __logf(s);
    s_tgt[q] = KCLS;
    const float thp = pb[q * 5 + 4];
    s_pc[q] = __cosf(thp);
    s_ps[q] = __sinf(thp);
  }
  __syncthreads();

  // ---- Phase B: cost matrix in 16x16 tiles; angle term via WMMA rank-4 GEMM ----
  // ang = ANG_W*(1 - cos(dq - dm)); cos(dq-dm) = [cos dq, sin dq] . [cos dm, sin dm]
  const int nQT = QPAD / 16;       // 57
  const int nMT = MN / 16;         // 8
  const int nT  = nQT * nMT;       // 456 (divisible by 8 waves -> uniform trip count)
  for (int t = wave; t < nT; t += NTHREADS / 32) {
    const int qt = t / nMT, mt = t % nMT;
    const int q0 = qt * 16, m0 = mt * 16;
    const bool lo = lane < 16;
    const int ml = m0 + (lane & 15);

    // A (16x4): lanes 0-15 hold {K0,K1} = {cos,sin} of row q0+lane; lanes 16-31 {K2,K3}=0
    int qa = q0 + lane;
    qa = qa < QN ? qa : QN - 1;
    v2f a;
    a.x = lo ? s_pc[qa] : 0.f;
    a.y = lo ? s_ps[qa] : 0.f;

    // B (4x16): v0 lanes0-15 = row K0 (cos over N), v1 lanes0-15 = row K1 (sin); upper = 0
    v2f b;
    b.x = lo ? s_tc[ml] : 0.f;
    b.y = lo ? s_ts[ml] : 0.f;

    v8f acc = {};
    acc = __builtin_amdgcn_wmma_f32_16x16x4_f32(
        /*neg_a=*/false, a, /*neg_b=*/false, b,
        /*c_mod=*/(short)0, acc, /*reuse_a=*/false, /*reuse_b=*/false);

    // C/D layout: reg r, lane l -> (q = q0 + r + 8*(l>=16), m = m0 + (l&15))
    const float* nm = s_nb + ml * 5;
    const int labm = s_lab[ml];
    const float n0 = nm[0], n1 = nm[1], n2 = nm[2], n3 = nm[3];
    const int qbase = q0 + ((lane >> 4) << 3);
    #pragma unroll
    for (int r = 0; r < 8; ++r) {
      const int q  = qbase + r;
      const int qc = q < QN ? q : QN - 1;   // clamp loads only; store to padded row is fine
      const float* pq = pb + qc * 5;
      const float l1 = fabsf(pq[0] - n0) + fabsf(pq[1] - n1) +
                       fabsf(pq[2] - n2) + fabsf(pq[3] - n3);
      const float cls = -CLS_W * __expf(lg[qc * KP1 + labm] - s_lse[qc]);
      const float ang = ANG_W * (1.f - acc[r]);
      cost[q * MN + ml] = cls + BBOX_W * l1 + ang;
    }
  }
  __threadfence_block();
  __syncthreads();

  // ---- Phase C: greedy matching with cached per-row minima ----
  for (int q = tid; q < QN; q += NTHREADS) {
    const float* row = cost + q * MN;
    float mn = BIGF; int am = 0;
    for (int m = 0; m < MN; ++m) {
      const float v = row[m];
      if (v < mn) { mn = v; am = m; }
    }
    s_rowmin[q] = mn; s_rowarg[q] = am;
  }
  __syncthreads();

  for (int it = 0; it < MN; ++it) {
    // block argmin over rowmin (tie-break: smallest q, matching row-major argmin)
    float bv = BIGF; int bi = QN;
    for (int q = tid; q < QN; q += NTHREADS) {
      const float v = s_rowmin[q];
      if (v < bv || (v == bv && q < bi)) { bv = v; bi = q; }
    }
    s_rv[tid] = bv; s_ri[tid] = bi;
    __syncthreads();
    for (int off = NTHREADS / 2; off > 0; off >>= 1) {
      if (tid < off) {
        const float ov = s_rv[tid + off]; const int oi = s_ri[tid + off];
        if (ov < s_rv[tid] || (ov == s_rv[tid] && oi < s_ri[tid])) {
          s_rv[tid] = ov; s_ri[tid] = oi;
        }
      }
      __syncthreads();
    }
    const int qstar = s_ri[0];
    const int mstar = s_rowarg[qstar];
    if (tid == 0) {
      s_pq[it] = qstar; s_pm[it] = mstar;
      s_rowmin[qstar] = BIGF;
      s_used[mstar >> 5] |= (1u << (mstar & 31));
    }
    __syncthreads();
    // rescan only rows whose cached argmin column was consumed (L2-resident reads)
    for (int q = tid; q < QN; q += NTHREADS) {
      if (s_rowmin[q] < BIGF && s_rowarg[q] == mstar) {
        const float* row = cost + q * MN;
        float mn = BIGF; int am = -1;
        for (int m = 0; m < MN; ++m) {
          if (!((s_used[m >> 5] >> (m & 31)) & 1u)) {
            const float v = row[m];
            if (v < mn) { mn = v; am = m; }
          }
        }
        s_rowmin[q] = mn; s_rowarg[q] = am;
      }
    }
    __syncthreads();
  }

  // ---- Phase D: losses ----
  if (tid < MN) s_tgt[s_pq[tid]] = s_lab[s_pm[tid]];
  __syncthreads();

  float wn = 0.f, wsum = 0.f;
  for (int q = tid; q < QN; q += NTHREADS) {
    const int c = s_tgt[q];
    const float nll = s_lse[q] - lg[q * KP1 + c];
    const float w = (c == KCLS) ? NOOBJ_W : 1.f;
    wn += w * nll; wsum += w;
  }
  const float sum_wn = block_sum(wn, s_rv, tid);
  const float sum_w  = block_sum(wsum, s_rv, tid);

  float sb = 0.f, sa = 0.f;
  if (tid < MN) {
    const int q = s_pq[tid], m = s_pm[tid];
    const float* pq = pb + q * 5;
    const float* nm = s_nb + m * 5;
    sb = fabsf(pq[0] - nm[0]) + fabsf(pq[1] - nm[1]) +
         fabsf(pq[2] - nm[2]) + fabsf(pq[3] - nm[3]);
    sa = 1.f - __cosf(pq[4] - nm[4]);
  }
  const float sum_b = block_sum(sb, s_rv, tid);
  const float sum_a = block_sum(sa, s_rv, tid);

  if (tid == 0) {
    loss_ws[img * 3 + 0] = sum_wn / sum_w;
    loss_ws[img * 3 + 1] = BBOX_W * sum_b / (float)(MN * 4);
    loss_ws[img * 3 + 2] = ANG_W * sum_a / (float)MN;
  }
}

__global__ void finalize_kernel(const float* __restrict__ loss_ws,
                                float* __restrict__ out, int B) {
  const int tid = threadIdx.x;
  __shared__ float s[3];
  if (tid < 3) {
    float acc = 0.f;
    for (int i = 0; i < B; ++i) acc += loss_ws[i * 3 + tid];
    s[tid] = acc / (float)B;
  }
  __syncthreads();
  if (tid == 0) {
    out[0] = s[0];
    out[1] = s[1];
    out[2] = s[2];
    out[3] = s[0] + s[1] + s[2];
  }
}

extern "C" void kernel_launch(void* const* d_in, const int* in_sizes, int n_in,
                              void* d_out, int out_size, void* d_ws, size_t ws_size,
                              hipStream_t stream) {
  const float* logits = (const float*)d_in[0];  // [B,900,81]
  const float* pboxes = (const float*)d_in[1];  // [B,900,5]
  const float* tboxes = (const float*)d_in[2];  // [B,128,5]
  const float* sizes  = (const float*)d_in[3];  // [B,2]
  const int*   labels = (const int*)d_in[4];    // [B,128]
  float* out = (float*)d_out;

  const int B = in_sizes[3] / 2;

  float* cost = (float*)d_ws;                       // B*QPAD*128 floats
  float* loss = cost + (size_t)B * QPAD * MN;       // B*3 floats

  oriented_criterion_kernel<<<B, NTHREADS, 0, stream>>>(
      logits, pboxes, tboxes, sizes, labels, cost, loss);
  finalize_kernel<<<1, 32, 0, stream>>>(loss, out, B);
}